// RoIHeads_83133386982245
// MI455X (gfx1250) — compile-verified
//
#include <hip/hip_runtime.h>
#include <stdint.h>

// ---------------- problem constants (from reference) ----------------
#define NPROP      2000
#define NPAD       2048          // next pow2 for bitonic sort
#define NCLS       81            // classes incl. background
#define NFG        (NCLS - 1)    // 80 foreground classes
#define NDET       100
#define IMG_WH     800.0f
#define SCORE_TH   0.05f
#define NMS_TH     0.5f
#define MIN_SIZE   1.0f
#define BBOX_CLIP  4.135166556742356f   // log(1000/16)

typedef uint32_t v4u __attribute__((ext_vector_type(4)));
typedef uint32_t v8u __attribute__((ext_vector_type(8)));

// ---------------- helpers ----------------
__device__ __forceinline__ uint32_t float_to_ordered(float s) {
  uint32_t b = __float_as_uint(s);
  return (b & 0x80000000u) ? ~b : (b | 0x80000000u);
}
__device__ __forceinline__ float ordered_to_float(uint32_t u) {
  return (u & 0x80000000u) ? __uint_as_float(u & 0x7FFFFFFFu)
                           : __uint_as_float(~u);
}

// proposals come from LDS (TDM-staged); deltas gathered from global (L2-resident)
__device__ __forceinline__ void decode_box(const float* __restrict__ box_reg,
                                           const float* prop_lds,
                                           int i, int cls,
                                           float& x1, float& y1,
                                           float& x2, float& y2) {
  const float* dl = box_reg + (size_t)i * (NCLS * 4) + (size_t)cls * 4;
  float d0 = dl[0], d1 = dl[1], d2 = dl[2], d3 = dl[3];
  const float* pr = prop_lds + i * 4;
  float px1 = pr[0], py1 = pr[1], px2 = pr[2], py2 = pr[3];
  float w  = px2 - px1, h = py2 - py1;
  float cx = px1 + 0.5f * w, cy = py1 + 0.5f * h;
  float dx = d0 * 0.1f, dy = d1 * 0.1f;                       // / REG_W
  float dw = fminf(d2 * 0.2f, BBOX_CLIP);
  float dh = fminf(d3 * 0.2f, BBOX_CLIP);
  float pcx = dx * w + cx, pcy = dy * h + cy;
  float pw = expf(dw) * w, ph = expf(dh) * h;
  x1 = fminf(fmaxf(pcx - 0.5f * pw, 0.0f), IMG_WH);
  y1 = fminf(fmaxf(pcy - 0.5f * ph, 0.0f), IMG_WH);
  x2 = fminf(fmaxf(pcx + 0.5f * pw, 0.0f), IMG_WH);
  y2 = fminf(fmaxf(pcy + 0.5f * ph, 0.0f), IMG_WH);
}

// ---------------- kernel 1: softmax -> transposed class scores ----------------
// ws_scores layout: [NFG][NPROP], class c (1..80) at row (c-1). Coalesced stores.
__global__ void softmax_scores_kernel(const float* __restrict__ logit,
                                      float* __restrict__ ws_scores) {
  int i = blockIdx.x * blockDim.x + threadIdx.x;
  if (i >= NPROP) return;
  const float* row = logit + (size_t)i * NCLS;
  __builtin_prefetch(row, 0, 0);                 // gfx1250 global_prefetch_b8
  float m = -3.4e38f;
  for (int c = 0; c < NCLS; ++c) m = fmaxf(m, row[c]);
  float sum = 0.0f;
  for (int c = 0; c < NCLS; ++c) sum += expf(row[c] - m);
  float inv = 1.0f / sum;
  for (int c = 1; c < NCLS; ++c)
    ws_scores[(size_t)(c - 1) * NPROP + i] = expf(row[c] - m) * inv;
}

// ---------------- kernel 2: per-class decode + sort + NMS + top-k -------------
// One 1024-thread (32-wave) workgroup per foreground class; whole working set in LDS.
__global__ void __launch_bounds__(1024)
nms_class_kernel(const float* __restrict__ box_reg,
                 const float* __restrict__ proposal,
                 const float* __restrict__ ws_scores,
                 float* __restrict__ out) {
  const int tid = threadIdx.x;
  const int c   = blockIdx.x;        // 0..79
  const int cls = c + 1;             // class id 1..80

  __shared__ __align__(16) float sp[NPROP * 4];       // TDM-staged proposals (32 KB)
  __shared__ __align__(16) float ls_score[NPROP];     // async-staged class scores
  __shared__ unsigned long long keys[NPAD];           // (ordered score)<<32 | (2047-i)
  __shared__ float bx1[NPROP], by1[NPROP], bx2[NPROP], by2[NPROP];
  __shared__ unsigned char keep[NPAD];
  __shared__ int  sh_sel[NDET];
  __shared__ int  sh_nvalid;

  if (tid == 0) sh_nvalid = 0;

  // ---- stage 0a: TDM DMA of the whole proposal table into LDS (wave 0 only;
  //      tensor ops ignore EXEC and are per-wave, so exactly one DMA is issued).
  //      D#: 1-D tile, 8000 x 4B elements, tracked on TENSORcnt.
  if (tid < 32) {
    uint32_t lds  = (uint32_t)(uintptr_t)(&sp[0]);     // low 32 bits = LDS offset
    uint64_t ga   = (uint64_t)(uintptr_t)proposal;
    v4u g0;
    g0.x = 1u;                                         // count=1, user mode
    g0.y = lds;                                        // lds_addr
    g0.z = (uint32_t)ga;                               // global_addr[31:0]
    g0.w = (uint32_t)((ga >> 32) & 0x01FFFFFFu) | (2u << 30);  // ga[56:32] | type=2
    v8u g1;
    g1.s0 = (2u << 16);                                // wg_mask=0, data_size=4B
    g1.s1 = (uint32_t)(NPROP * 4) << 16;               // tensor_dim0 = 8000 (lo16)
    g1.s2 = (1u << 16);                                // tensor_dim0 hi=0, tensor_dim1=1
    g1.s3 = (uint32_t)(NPROP * 4) << 16;               // tile_dim0 = 8000
    g1.s4 = 1u;                                        // tile_dim1=1, tile_dim2=0
    g1.s5 = (uint32_t)(NPROP * 4);                     // tensor_dim0_stride = 8000
    g1.s6 = 0u;
    g1.s7 = 0u;
    asm volatile("tensor_load_to_lds %0, %1" :: "s"(g0), "s"(g1) : "memory");
  }

  // ---- stage 0b: async global -> LDS staging of this class's 2000 scores ----
  // 500 x b128 transfers, tracked on ASYNCcnt; overlaps with the TDM transfer.
  {
    const float* gsc = ws_scores + (size_t)c * NPROP;       // 8000B offset: 16B aligned
    uint32_t lbase = (uint32_t)(uintptr_t)(&ls_score[0]);
    for (int r = tid; r < NPROP / 4; r += 1024) {
      uint32_t laddr = lbase + (uint32_t)(r * 16);
      unsigned long long gaddr = (unsigned long long)(uintptr_t)(gsc + r * 4);
      asm volatile("global_load_async_to_lds_b128 %0, %1, off"
                   :: "v"(laddr), "v"(gaddr) : "memory");
    }
    asm volatile("s_wait_asynccnt 0x0" ::: "memory");
  }
  if (tid < 32) __builtin_amdgcn_s_wait_tensorcnt(0);   // drain TDM in issuing wave
  __syncthreads();                                      // publish LDS to all waves

  // ---- stage 1: decode + filter, build sort keys ----
  for (int i = tid; i < NPAD; i += 1024) {
    float s;
    if (i < NPROP) {
      float x1, y1, x2, y2;
      decode_box(box_reg, sp, i, cls, x1, y1, x2, y2);
      float bw = x2 - x1, bh = y2 - y1;
      float sc = ls_score[i];
      bool valid = (sc >= SCORE_TH) && (bw >= MIN_SIZE) && (bh >= MIN_SIZE);
      s = valid ? sc : -1.0f;
    } else {
      s = -3.0f;                       // padding sorts strictly after the -1 tail
    }
    uint32_t u = float_to_ordered(s);
    keys[i] = ((unsigned long long)u << 32) |
              (unsigned long long)(NPAD - 1 - i);   // reversed idx -> stable ties
  }

  // ---- stage 2: bitonic sort, descending, 2048 keys / 1024 threads ----
  for (unsigned k = 2; k <= NPAD; k <<= 1) {
    for (unsigned j = k >> 1; j > 0; j >>= 1) {
      __syncthreads();
      for (unsigned i = tid; i < NPAD; i += 1024) {
        unsigned ixj = i ^ j;
        if (ixj > i) {
          bool dir = ((i & k) == 0);               // descending block
          unsigned long long a = keys[i], b = keys[ixj];
          if (dir ? (a < b) : (a > b)) { keys[i] = b; keys[ixj] = a; }
        }
      }
    }
  }
  __syncthreads();

  // ---- stage 3: gather sorted boxes (re-decode by original index, LDS proposals) ----
  for (int r = tid; r < NPROP; r += 1024) {
    unsigned long long key = keys[r];
    int idx = NPAD - 1 - (int)(key & 0xFFFFFFFFu);
    float s = ordered_to_float((uint32_t)(key >> 32));
    if (idx < NPROP) {
      float x1, y1, x2, y2;
      decode_box(box_reg, sp, idx, cls, x1, y1, x2, y2);
      bx1[r] = x1; by1[r] = y1; bx2[r] = x2; by2[r] = y2;
      unsigned char kp = (s >= 0.0f) ? 1 : 0;
      keep[r] = kp;
      if (kp) atomicAdd(&sh_nvalid, 1);
    }
  }
  __syncthreads();

  // ---- stage 4: greedy NMS; serial over pivots, parallel over victims ----
  const int nv = sh_nvalid;            // invalid entries sorted to the tail
  for (int i = 0; i < nv; ++i) {
    if (!keep[i]) continue;            // block-uniform (LDS broadcast), skip barrier
    float x1 = bx1[i], y1 = by1[i], x2 = bx2[i], y2 = by2[i];
    float ai = (x2 - x1) * (y2 - y1);
    for (int j = i + 1 + tid; j < nv; j += 1024) {
      if (keep[j]) {
        float xx1 = fmaxf(x1, bx1[j]);
        float yy1 = fmaxf(y1, by1[j]);
        float xx2 = fminf(x2, bx2[j]);
        float yy2 = fminf(y2, by2[j]);
        float iw = fmaxf(xx2 - xx1, 0.0f);
        float ih = fmaxf(yy2 - yy1, 0.0f);
        float inter = iw * ih;
        float aj = (bx2[j] - bx1[j]) * (by2[j] - by1[j]);
        float iou = inter / (ai + aj - inter + 1e-9f);
        if (iou > NMS_TH) keep[j] = 0;
      }
    }
    __syncthreads();
  }
  __syncthreads();

  // ---- stage 5: stable top-100 (matches jax.lax.top_k tie behavior) ----
  // final[r] = keep[r] ? score : -1 ; kept entries already rank-ordered,
  // remaining slots fill with smallest sorted positions having final == -1.
  if (tid == 0) {
    int cnt = 0;
    for (int r = 0; r < NPROP && cnt < NDET; ++r) if (keep[r])  sh_sel[cnt++] = r;
    for (int r = 0; r < NPROP && cnt < NDET; ++r) if (!keep[r]) sh_sel[cnt++] = r;
  }
  __syncthreads();

  // ---- stage 6: emit boxes / scores / labels / valid ----
  float* out_boxes  = out;                           // [8000][4]
  float* out_scores = out + NFG * NDET * 4;          // [8000]
  float* out_labels = out + NFG * NDET * 5;          // [8000]
  float* out_valid  = out + NFG * NDET * 6;          // [8000]
  for (int k = tid; k < NDET; k += 1024) {
    int r = sh_sel[k];
    bool kp = keep[r] != 0;
    float sc = kp ? ordered_to_float((uint32_t)(keys[r] >> 32)) : -1.0f;
    int o = c * NDET + k;
    out_boxes[o * 4 + 0] = bx1[r];
    out_boxes[o * 4 + 1] = by1[r];
    out_boxes[o * 4 + 2] = bx2[r];
    out_boxes[o * 4 + 3] = by2[r];
    out_scores[o] = sc;
    out_labels[o] = (float)cls;
    out_valid[o]  = kp ? 1.0f : 0.0f;
  }
}

// ---------------- launch ----------------
extern "C" void kernel_launch(void* const* d_in, const int* in_sizes, int n_in,
                              void* d_out, int out_size, void* d_ws, size_t ws_size,
                              hipStream_t stream) {
  const float* class_logit = (const float*)d_in[0];   // [2000, 81]
  const float* box_reg     = (const float*)d_in[1];   // [2000, 324]
  const float* proposal    = (const float*)d_in[2];   // [2000, 4]
  float* out       = (float*)d_out;
  float* ws_scores = (float*)d_ws;                    // [80][2000] floats = 640 KB

  softmax_scores_kernel<<<(NPROP + 255) / 256, 256, 0, stream>>>(class_logit, ws_scores);
  nms_class_kernel<<<NFG, 1024, 0, stream>>>(box_reg, proposal, ws_scores, out);
}